// Decoder_52398601011370
// MI455X (gfx1250) — compile-verified
//
#include <hip/hip_runtime.h>

// CDNA5 / gfx1250: wave32, WMMA (not MFMA).
typedef __attribute__((ext_vector_type(2))) float v2f;
typedef __attribute__((ext_vector_type(8))) float v8f;

#define B_DIM   2048
#define A_DIM   512
#define NABS_D  16
#define E_DIM   256
#define H_DIM   256
#define OUT_DIM 2

// ---------------------------------------------------------------------------
// Kernel 1: emb_base[a][h] = b1[h] + sum_e embed[a][e] * W1[1+e][h]
// Exact fp32 GEMM via V_WMMA_F32_16X16X4_F32. M=512, N=256, K=256.
// One wave per 16x16 output tile; 64 WMMAs per wave (K in chunks of 4).
// ---------------------------------------------------------------------------
__global__ __launch_bounds__(128) void precompute_emb_kernel(
    const float* __restrict__ embed,    // [512, 256]
    const float* __restrict__ W1,       // [257, 256]
    const float* __restrict__ b1,       // [256]
    float* __restrict__ embBase)        // [512, 256] (workspace)
{
    const int lane = threadIdx.x & 31;
    const int wave = threadIdx.x >> 5;
    const int tile = blockIdx.x * 4 + wave;   // 512 tiles total
    const int mt   = tile >> 4;               // 0..31 : a-tile
    const int nt   = tile & 15;               // 0..15 : h-tile
    const int n    = lane & 15;               // A row index == B/C col index
    const int half = lane >> 4;

    v8f acc = {};
    for (int kb = 0; kb < E_DIM; kb += 4) {
        const int kk = kb + 2 * half;
        // A (16x4 f32): lane holds row m=n, K = kk, kk+1  (contiguous -> b64 load)
        v2f afrag = *(const v2f*)(embed + (size_t)(mt * 16 + n) * E_DIM + kk);
        // B (4x16 f32): lane holds col n, K = kk, kk+1 (stride H apart in memory)
        v2f bfrag;
        bfrag.x = W1[(size_t)(1 + kk) * H_DIM + nt * 16 + n];
        bfrag.y = W1[(size_t)(2 + kk) * H_DIM + nt * 16 + n];
        acc = __builtin_amdgcn_wmma_f32_16x16x4_f32(
                  false, afrag, false, bfrag, (short)0, acc, false, false);
    }

    const float bias = b1[nt * 16 + n];
#pragma unroll
    for (int i = 0; i < 8; ++i) {
        const int mrow = i + 8 * half;  // C/D: VGPR i -> M=i (lanes 0-15), M=8+i (16-31)
        embBase[(size_t)(mt * 16 + mrow) * H_DIM + nt * 16 + n] = acc[i] + bias;
    }
}

// ---------------------------------------------------------------------------
// Kernel 2: out[b, a0+m, o] = b2[o] + sum_h relu(s*u_h + v_{a,h}) * W2[h,o]
//   s = abs_actions[b, assign[b,a]],  u = W1[0,:],  v = emb_base (from ws).
// One wave handles a 16-a tile (WMMA M dim) for NB=8 batches, W2 padded to
// N=16. v-tile staged in LDS with +4 float row skew (4-bank step per row ->
// conflict-free ds_load_b64 across the 16 M-lanes).
// ---------------------------------------------------------------------------
#define WAVES 4
#define NB    8
#define VSTR  (H_DIM + 4)   // 260 floats: 1040B row stride = 16B mod 256 -> skew

__global__ __launch_bounds__(128) void decoder_main_kernel(
    const float* __restrict__ absAct,   // [2048, 16]
    const long long* __restrict__ asg,  // [2048, 512] int64
    const float* __restrict__ W1,       // row 0 = u [256]
    const float* __restrict__ W2,       // [256, 2]
    const float* __restrict__ b2,       // [2]
    const float* __restrict__ embBase,  // [512, 256] (workspace)
    float* __restrict__ out)            // [2048, 512, 2]
{
    __shared__ __align__(16) float vT[16 * VSTR];
    __shared__ __align__(16) float uR[H_DIM];
    __shared__ __align__(16) float w2s[H_DIM * OUT_DIM];

    const int tid = threadIdx.x;
    const int a0  = blockIdx.y * 16;

    // Stage the 16-row emb_base tile, u, and W2 into LDS.
    for (int t = tid; t < 16 * H_DIM; t += 128) {
        const int r = t >> 8, c = t & (H_DIM - 1);
        vT[r * VSTR + c] = embBase[(size_t)(a0 + r) * H_DIM + c];
    }
    for (int t = tid; t < H_DIM; t += 128)            uR[t]  = W1[t];
    for (int t = tid; t < H_DIM * OUT_DIM; t += 128)  w2s[t] = W2[t];
    __syncthreads();

    const int lane  = tid & 31;
    const int wave  = tid >> 5;
    const int n     = lane & 15;   // A-row (a offset) == B/C column
    const int half  = lane >> 4;
    const int bBase = (blockIdx.x * WAVES + wave) * NB;

    // Gather the 16 assigned scalars for each of NB batches (int64 indices).
    float s[NB];
#pragma unroll
    for (int j = 0; j < NB; ++j) {
        const int b   = bBase + j;
        const int idx = (int)asg[(size_t)b * A_DIM + a0 + n];
        s[j] = absAct[b * NABS_D + idx];
    }

    v8f acc[NB] = {};

    for (int kb = 0; kb < H_DIM; kb += 4) {
        const int kk = kb + 2 * half;
        const v2f u = *(const v2f*)(uR + kk);                 // broadcast read
        const v2f v = *(const v2f*)(vT + n * VSTR + kk);      // skewed, b64
        v2f w;                                                // B: W2 padded N=16
        w.x = (n < OUT_DIM) ? w2s[kk * OUT_DIM + n] : 0.0f;
        w.y = (n < OUT_DIM) ? w2s[(kk + 1) * OUT_DIM + n] : 0.0f;
#pragma unroll
        for (int j = 0; j < NB; ++j) {
            v2f h;
            h.x = fmaxf(fmaf(s[j], u.x, v.x), 0.0f);
            h.y = fmaxf(fmaf(s[j], u.y, v.y), 0.0f);
            acc[j] = __builtin_amdgcn_wmma_f32_16x16x4_f32(
                         false, h, false, w, (short)0, acc[j], false, false);
        }
    }

    // Only columns 0,1 of each C tile are real outputs.
    if (n < OUT_DIM) {
        const float bias = b2[n];
#pragma unroll
        for (int j = 0; j < NB; ++j) {
            const int b = bBase + j;
#pragma unroll
            for (int i = 0; i < 8; ++i) {
                const int mrow = i + 8 * half;
                out[((size_t)b * A_DIM + a0 + mrow) * OUT_DIM + n] = acc[j][i] + bias;
            }
        }
    }
}

// ---------------------------------------------------------------------------
extern "C" void kernel_launch(void* const* d_in, const int* in_sizes, int n_in,
                              void* d_out, int out_size, void* d_ws, size_t ws_size,
                              hipStream_t stream) {
    // d_in order: state(0, unused), abs_actions(1), assignments(2, int64),
    //             embed_table(3), W1(4), b1(5), W2(6), b2(7)
    const float*     absAct = (const float*)d_in[1];
    const long long* asg    = (const long long*)d_in[2];
    const float*     embed  = (const float*)d_in[3];
    const float*     W1     = (const float*)d_in[4];
    const float*     b1     = (const float*)d_in[5];
    const float*     W2     = (const float*)d_in[6];
    const float*     b2     = (const float*)d_in[7];
    float*           outp   = (float*)d_out;
    float*           embBase = (float*)d_ws;   // 512*256 f32 = 512 KB scratch

    // Phase 1: batch-independent emb_base = embed @ W1[1:] + b1 (WMMA fp32).
    precompute_emb_kernel<<<128, 128, 0, stream>>>(embed, W1, b1, embBase);

    // Phase 2: gather + relu + padded-N WMMA contraction.
    // grid.x = 2048 / (4 waves * 8 b) = 64 ; grid.y = 512 / 16 = 32 a-tiles.
    decoder_main_kernel<<<dim3(64, 32), 128, 0, stream>>>(
        absAct, asg, W1, W2, b2, embBase, outp);
}